// GCNEncoder_62440234549675
// MI455X (gfx1250) — compile-verified
//
#include <hip/hip_runtime.h>
#include <math.h>

// ---------------- problem constants (from reference) ----------------
#define NN     50000   // nodes
#define EE     800000  // edges
#define DD     100     // embedding dim
#define EDIM   20      // edge dim
#define DCAT   120     // D + ED
#define NTILES 7       // ceil(100/16)
#define MTILES 3125    // 50000/16 exactly

typedef __attribute__((ext_vector_type(2))) float v2f;
typedef __attribute__((ext_vector_type(8))) float v8f;

// ---------------- one-time prep kernels ----------------

// deg accumulator init: every node has a self loop -> deg starts at 1
__global__ void k_deg_init(float* deg) {
    int i = blockIdx.x * blockDim.x + threadIdx.x;
    if (i < NN) deg[i] = 1.0f;
}

// deg += count of occurrences as source (row) over real edges
__global__ void k_deg_edges(const int* __restrict__ row, float* deg) {
    int e = blockIdx.x * blockDim.x + threadIdx.x;
    if (e < EE) atomicAdd(&deg[row[e]], 1.0f);
}

// dinv = rsqrt(deg)  (deg >= 1 always, no zero guard needed)
__global__ void k_dinv(float* deg_to_dinv) {
    int i = blockIdx.x * blockDim.x + threadIdx.x;
    if (i < NN) deg_to_dinv[i] = rsqrtf(deg_to_dinv[i]);
}

// max_norm=1 embedding lookup: X[i,:] = emb[x_idx[i],:] * min(1, 1/max(||.||,1e-7))
// one wave32 per node; lanes 0..24 each own one float4 (25*4 = 100 elements)
__global__ void k_maxnorm(const float* __restrict__ emb, const int* __restrict__ x_idx,
                          float* __restrict__ X) {
    int wave = (blockIdx.x * blockDim.x + threadIdx.x) >> 5;
    int lane = threadIdx.x & 31;
    if (wave >= NN) return;
    const float* src = emb + (size_t)x_idx[wave] * DD;   // 400B-aligned rows
    float4 v = make_float4(0.f, 0.f, 0.f, 0.f);
    if (lane < 25) v = *(const float4*)(src + 4 * lane);
    float s = v.x * v.x + v.y * v.y + v.z * v.z + v.w * v.w;
    for (int off = 16; off > 0; off >>= 1) s += __shfl_xor(s, off, 32);
    float nrm = sqrtf(s);
    float scale = fminf(1.0f, 1.0f / fmaxf(nrm, 1e-7f));
    if (lane < 25) {
        float4 o = make_float4(v.x * scale, v.y * scale, v.z * scale, v.w * scale);
        *(float4*)(X + (size_t)wave * DD + 4 * lane) = o;
    }
}

// ---------------- WMMA GEMM:  C[NN x 100] = A[NN x K] @ B[K x 100] (+bias, +ELU) ----
// V_WMMA_F32_16X16X4_F32, fp32 in/acc (exact fp32 semantics, matches reference).
// K templated (100 or 120) -> fully unrolled; B column slab cached in registers and
// reused across MT m-tiles per wave. Padded columns n=100..111 use a CLAMPED column
// index (in-bounds loads, garbage results, never stored) so every load is
// unconditional -> no EXEC juggling around the WMMAs.
template <int K, int MT>
__global__ void k_gemm_wmma(const float* __restrict__ A,
                            const float* __restrict__ B,
                            const float* __restrict__ bias,
                            float* __restrict__ C, int do_elu) {
    constexpr int KSTEPS  = K / 4;
    constexpr int MGROUPS = (MTILES + MT - 1) / MT;

    int wavesPerBlk = blockDim.x >> 5;
    int w = blockIdx.x * wavesPerBlk + (threadIdx.x >> 5);
    if (w >= MGROUPS * NTILES) return;            // wave-uniform
    int mg    = w / NTILES;
    int ntile = w - mg * NTILES;

    int lane  = threadIdx.x & 31;
    int idx16 = lane & 15;                        // A row index / B,C col index
    int half  = lane >> 4;                        // selects K pair (0-1 vs 2-3)
    int n     = ntile * 16 + idx16;
    bool nok  = (n < DD);
    int nc    = nok ? n : (DD - 1);               // clamped column for loads

    // stage B column slab in registers, reused across MT m-tiles
    v2f breg[KSTEPS];
#pragma unroll
    for (int ks = 0; ks < KSTEPS; ++ks) {
        int kk = ks * 4 + half * 2;
        breg[ks].x = B[(size_t)kk * DD + nc];
        breg[ks].y = B[(size_t)(kk + 1) * DD + nc];
    }
    float bn = (bias != nullptr) ? bias[nc] : 0.0f;

    for (int mt = 0; mt < MT; ++mt) {
        int mtile = mg * MT + mt;
        if (mtile >= MTILES) break;               // wave-uniform
        const float* arow = A + (size_t)(mtile * 16 + idx16) * K + half * 2;
        v8f acc = {};
#pragma unroll
        for (int ks = 0; ks < KSTEPS; ++ks) {
            v2f a = *(const v2f*)(arow + ks * 4); // 8B-aligned contiguous pair
            acc = __builtin_amdgcn_wmma_f32_16x16x4_f32(
                /*neg_a=*/false, a, /*neg_b=*/false, breg[ks],
                /*c_mod=*/(short)0, acc, /*reuse_a=*/false, /*reuse_b=*/false);
        }
        if (nok) {
#pragma unroll
            for (int v = 0; v < 8; ++v) {
                int r = mtile * 16 + v + 8 * half; // C/D layout: row = v + 8*half
                float val = acc[v] + bn;
                if (do_elu) val = (val > 0.0f) ? val : (expf(val) - 1.0f);
                C[(size_t)r * DD + n] = val;
            }
        }
    }
}

// ---------------- aggr init with self-loop contribution ----------------
// aggr[i, 0:100]  = dinv[i]^2 * H[i, :]; aggr[i,100:120] = 0. One float4 per thread.
__global__ void k_init_aggr(const float* __restrict__ H, const float* __restrict__ dinv,
                            float* __restrict__ AGG) {
    size_t t = (size_t)blockIdx.x * blockDim.x + threadIdx.x;
    size_t total = (size_t)NN * (DCAT / 4);       // 30 float4 per node
    if (t >= total) return;
    size_t i = t / (DCAT / 4);
    int    q = (int)(t - i * (DCAT / 4));
    float4 o = make_float4(0.f, 0.f, 0.f, 0.f);
    if (q < DD / 4) {
        float4 h4 = *(const float4*)(H + i * DD + 4 * q);
        float s = dinv[i] * dinv[i];
        o = make_float4(s * h4.x, s * h4.y, s * h4.z, s * h4.w);
    }
    *(float4*)(AGG + i * DCAT + 4 * q) = o;
}

// ---------------- edge scatter: aggr[col] += norm * [H[row] , ea] ----------------
// one wave32 per edge (grid-stride); lanes 0..24 carry H[row] (float4 each),
// lanes 25..29 carry edge_emb (float4 each), lanes 30/31 idle.
__global__ void k_edge_scatter(const int* __restrict__ row, const int* __restrict__ col,
                               const int* __restrict__ eattr,
                               const float* __restrict__ edge_emb,
                               const float* __restrict__ H,
                               const float* __restrict__ dinv,
                               float* __restrict__ AGG) {
    int wave   = (blockIdx.x * blockDim.x + threadIdx.x) >> 5;
    int lane   = threadIdx.x & 31;
    int nwaves = (gridDim.x * blockDim.x) >> 5;
    for (int e = wave; e < EE; e += nwaves) {
        int r = row[e];
        int c = col[e];
        float nrm = dinv[r] * dinv[c];
        float4 v = make_float4(0.f, 0.f, 0.f, 0.f);
        int j0 = 0;
        if (lane < 25) {
            v  = *(const float4*)(H + (size_t)r * DD + 4 * lane);
            j0 = 4 * lane;
        } else if (lane < 30) {
            v  = *(const float4*)(edge_emb + (size_t)eattr[e] * EDIM + 4 * (lane - 25));
            j0 = DD + 4 * (lane - 25);
        }
        if (lane < 30) {
            float* dst = AGG + (size_t)c * DCAT + j0;
            atomicAdd(dst + 0, nrm * v.x);
            atomicAdd(dst + 1, nrm * v.y);
            atomicAdd(dst + 2, nrm * v.z);
            atomicAdd(dst + 3, nrm * v.w);
        }
    }
}

// ---------------- driver ----------------
extern "C" void kernel_launch(void* const* d_in, const int* in_sizes, int n_in,
                              void* d_out, int out_size, void* d_ws, size_t ws_size,
                              hipStream_t stream) {
    const float* node_emb = (const float*)d_in[0];   // [NN,100]
    const float* edge_emb = (const float*)d_in[1];   // [64,20]
    const float* w1       = (const float*)d_in[2];   // [100,100]
    const float* ew1      = (const float*)d_in[3];   // [120,100]
    const float* b1       = (const float*)d_in[4];   // [100]
    const float* w2       = (const float*)d_in[5];
    const float* ew2      = (const float*)d_in[6];
    const float* b2       = (const float*)d_in[7];
    const int*   x_idx    = (const int*)d_in[8];     // [NN]
    const int*   e_row    = (const int*)d_in[9];     // edge_index[0]
    const int*   e_col    = e_row + EE;              // edge_index[1]
    const int*   e_attr   = (const int*)d_in[10];    // [EE]

    float* X    = (float*)d_ws;                      // [NN,100]  20 MB
    float* H    = X   + (size_t)NN * DD;             // [NN,100]  20 MB
    float* AGG  = H   + (size_t)NN * DD;             // [NN,120]  24 MB
    float* DINV = AGG + (size_t)NN * DCAT;           // [NN]      0.2 MB

    float* OUT = (float*)d_out;

    // ---- one-time prep ----
    k_deg_init <<<(NN + 255) / 256, 256, 0, stream>>>(DINV);
    k_deg_edges<<<(EE + 255) / 256, 256, 0, stream>>>(e_row, DINV);
    k_dinv     <<<(NN + 255) / 256, 256, 0, stream>>>(DINV);
    k_maxnorm  <<<(NN * 32 + 255) / 256, 256, 0, stream>>>(node_emb, x_idx, X);

    constexpr int MT = 4;                               // m-tiles per wave
    constexpr int MGROUPS = (MTILES + MT - 1) / MT;     // 782
    const int gemmBlk  = 128;                           // 4 waves/block
    const int gemmGrid = (MGROUPS * NTILES + 3) / 4;    // one wave per tile-group
    const int aggrGrid = (int)(((size_t)NN * (DCAT / 4) + 255) / 256);
    const int edgeGrid = 4096;                          // grid-stride, 32768 waves

    // ---- 3 rounds x 2 convs ----
    for (int round = 0; round < 3; ++round) {
        for (int phase = 0; phase < 2; ++phase) {
            const float* W  = phase ? w2  : w1;
            const float* EW = phase ? ew2 : ew1;
            const float* Bv = phase ? b2  : b1;
            bool last = (round == 2 && phase == 1);
            float* dst = last ? OUT : X;
            int elu = (phase == 0) ? 1 : 0;

            // h = x @ W   (K = 100)
            k_gemm_wmma<DD, MT><<<gemmGrid, gemmBlk, 0, stream>>>(X, W, nullptr, H, 0);
            // aggr = self-loop term (+ zero edge-feature tail)
            k_init_aggr<<<aggrGrid, 256, 0, stream>>>(H, DINV, AGG);
            // aggr += scattered edge messages
            k_edge_scatter<<<edgeGrid, 256, 0, stream>>>(e_row, e_col, e_attr,
                                                         edge_emb, H, DINV, AGG);
            // x = act(aggr @ EW + b)   (K = 120)
            k_gemm_wmma<DCAT, MT><<<gemmGrid, gemmBlk, 0, stream>>>(AGG, EW, Bv, dst, elu);
        }
    }
}